// MultiHeadAttentionQuantum_65481071399891
// MI455X (gfx1250) — compile-verified
//
#include <hip/hip_runtime.h>

typedef __attribute__((ext_vector_type(2))) float v2f;
typedef __attribute__((ext_vector_type(4))) float v4f;
typedef __attribute__((ext_vector_type(8))) float v8f;
typedef __attribute__((ext_vector_type(4))) unsigned int u32x4;
typedef __attribute__((ext_vector_type(4))) int i32x4;
typedef __attribute__((ext_vector_type(8))) int i32x8;

namespace {
constexpr int kB   = 2;
constexpr int kT   = 2048;
constexpr int kC   = 64;    // EMBED_DIM
constexpr int kH   = 32;    // NUM_HEADS
constexpr int kD   = 2;     // HEAD_DIM
constexpr int kTok = kB * kT;     // 4096 tokens
constexpr int kQKVC = 3 * kC;     // 192 qkv columns
constexpr int kKVSlice = 2 * kT * kD;  // 8192 f32: K-slice then V-slice of one head
}

// ---------------------------------------------------------------------------
// TDM: 1-row tensor DMA of n4 float32 elements global -> LDS.
// D# packed per CDNA5 ISA ch.8: group0 = {count=1 | lds_addr | global_addr |
// type=2}, group1 = {data_size=4B, tensor_dim0=tile_dim0=n4, tile_dim1=1,
// tensor_dim0_stride=n4}. Groups 2/3 zero (2D tile). cpol=0.
// ---------------------------------------------------------------------------
__device__ __forceinline__ void tdm_load_to_lds(unsigned int lds_off,
                                                const void* gsrc,
                                                unsigned int n4) {
  const unsigned long long ga = (unsigned long long)(uintptr_t)gsrc;
  u32x4 g0 = {1u,                                   // count=1 (valid user D#)
              lds_off,                              // lds_addr (bytes)
              (unsigned int)(ga & 0xFFFFFFFFu),     // global_addr[31:0]
              (unsigned int)((ga >> 32) & 0x01FFFFFFu) | 0x80000000u}; // +type=2
  i32x8 g1 = {(int)0x00020000u,                     // data_size=2 (4 bytes)
              (int)((n4 & 0xFFFFu) << 16),          // tensor_dim0[15:0]
              (int)(((n4 >> 16) & 0xFFFFu) | 0x10000u),  // dim0[31:16]|dim1=1
              (int)((n4 & 0xFFFFu) << 16),          // tile_dim0 = n4
              1,                                    // tile_dim1 = 1
              (int)n4,                              // tensor_dim0_stride[31:0]
              0, 0};
  i32x4 z4 = {0, 0, 0, 0};
#if defined(__clang_major__) && (__clang_major__ >= 23)
  i32x8 z8 = {0, 0, 0, 0, 0, 0, 0, 0};
  __builtin_amdgcn_tensor_load_to_lds(g0, g1, z4, z4, z8, 0);
#else
  __builtin_amdgcn_tensor_load_to_lds(g0, g1, z4, z4, 0);
#endif
}

// ---------------------------------------------------------------------------
// Kernel A: qkv = x @ W_qkv^T ; then q/k/v = cos(qkv) * cos(theta[d]).
// One wave = one 16x16 tile via chained f32 16x16x4 WMMAs (2 accumulators
// for WMMA ILP). Q scattered as [bh][t][d]; K,V interleaved per head as
// KV[bh*2 + {0:K,1:V}][t][d] so attention can DMA both with one descriptor.
// ---------------------------------------------------------------------------
__global__ __launch_bounds__(128) void qkv_rope_kernel(
    const float* __restrict__ x, const float* __restrict__ wqkv,
    const float* __restrict__ theta,
    float* __restrict__ Qo, float* __restrict__ KVo) {
  const int lane = threadIdx.x & 31;
  const int wv   = threadIdx.x >> 5;
  const int tile = blockIdx.x * 4 + wv;
  const int tilesN = kQKVC / 16;              // 12
  const int m0 = (tile / tilesN) * 16;
  const int n0 = (tile % tilesN) * 16;
  const int l15 = lane & 15;
  const int hi  = lane >> 4;                  // 0: K-dim 0..1, 1: K-dim 2..3

  const float* ap = x    + (m0 + l15) * kC + hi * 2;
  const float* bp = wqkv + (n0 + l15) * kC + hi * 2;

  v8f acc0 = {0.f, 0.f, 0.f, 0.f, 0.f, 0.f, 0.f, 0.f};
  v8f acc1 = {0.f, 0.f, 0.f, 0.f, 0.f, 0.f, 0.f, 0.f};
#pragma unroll
  for (int kk = 0; kk < kC; kk += 8) {
    v2f a0 = *(const v2f*)(ap + kk);
    v2f b0 = *(const v2f*)(bp + kk);
    v2f a1 = *(const v2f*)(ap + kk + 4);
    v2f b1 = *(const v2f*)(bp + kk + 4);
    acc0 = __builtin_amdgcn_wmma_f32_16x16x4_f32(false, a0, false, b0, (short)0,
                                                 acc0, false, false);
    acc1 = __builtin_amdgcn_wmma_f32_16x16x4_f32(false, a1, false, b1, (short)0,
                                                 acc1, false, false);
  }
#pragma unroll
  for (int i = 0; i < 8; ++i) acc0[i] += acc1[i];

  const int col = n0 + l15;
  const float ct = __cosf(theta[col & 1]);
  const int which = col >> 6;                 // 0=Q 1=K 2=V (uniform per tile)
  const int rem = col & (kC - 1);
  const int h = rem >> 1;
  const int d = rem & 1;

#pragma unroll
  for (int r = 0; r < 8; ++r) {
    const int row = m0 + r + hi * 8;          // global token index
    const int b = row >> 11;                  // / kT
    const int t = row & (kT - 1);
    const int bh = b * kH + h;
    const float val = __cosf(acc0[r]) * ct;
    if (which == 0) {
      Qo[((size_t)bh * kT + t) * kD + d] = val;
    } else {
      KVo[(((size_t)bh * 2 + (which - 1)) * kT + t) * kD + d] = val;
    }
  }
}

// ---------------------------------------------------------------------------
// Kernel B: streaming attention. One block (8 waves) = one (b,h) slice x 128
// queries. K+V slice (32 KB) staged to LDS once by the Tensor Data Mover;
// hot loop = ds_load_b64 (K frag) + v_wmma_f32_16x16x4_f32 (S^T tile) +
// 8x v_exp_f32 + ds_load_b128 V broadcast + FMAs, all in-register softmax.
// |scores| <= sqrt(2) (inputs are cos()*cos()), so exp cannot overflow and
// the unnormalized softmax (no running max) is exact.
// ---------------------------------------------------------------------------
__global__ __launch_bounds__(256) void flash_attn_kernel(
    const float* __restrict__ Qi, const float* __restrict__ KVi,
    float* __restrict__ Oo) {
  __shared__ float smem[kKVSlice];            // 8192 f32: K[2048][2], V[2048][2]

  const int lane = threadIdx.x & 31;
  const int wv   = threadIdx.x >> 5;
  const int bh   = blockIdx.x >> 4;           // 64 head-slices
  const int qt   = (blockIdx.x & 15) * 8 + wv;// 128 query tiles
  const int q0   = qt * 16;
  const int l15  = lane & 15;
  const int hi   = lane >> 4;
  const float hmask = (hi == 0) ? 1.0f : 0.0f;   // zero-pad K-dim rows 2,3

  // Stage K+V slice of this head into LDS with one TDM descriptor.
  if (wv == 0) {
    tdm_load_to_lds((unsigned int)(uintptr_t)(void*)smem,
                    KVi + (size_t)bh * kKVSlice, (unsigned int)kKVSlice);
    __builtin_amdgcn_s_wait_tensorcnt(0);
  }
  __syncthreads();
  const float* lK = smem;                     // keys:   [t][d]
  const float* lV = smem + kT * kD;           // values: [t][d]

  // B operand: Q^T * (1/sqrt(D)), loop invariant.
  v2f bq = *(const v2f*)(Qi + ((size_t)bh * kT + q0 + l15) * kD);
  bq *= (0.70710678118654752f * hmask);

  float l = 0.f, a0 = 0.f, a1 = 0.f;
  const v8f zero8 = {0.f, 0.f, 0.f, 0.f, 0.f, 0.f, 0.f, 0.f};

#pragma unroll 2
  for (int k0 = 0; k0 < kT; k0 += 16) {
    // A operand: 16 keys (M) x key-vector (K-dim 0..1), rows 2..3 zero.
    v2f ak = *(const v2f*)(lK + (k0 + l15) * kD);
    ak *= hmask;
    // st[r] = score(query = q0 + l15, key = k0 + r + hi*8)
    v8f st = __builtin_amdgcn_wmma_f32_16x16x4_f32(
        false, ak, false, bq, (short)0, zero8, false, false);

    const int kb = k0 + hi * 8;
    const v4f* vp = (const v4f*)(lV + kb * kD);  // 8 keys * (v0,v1), 64B aligned
    const v4f u0 = vp[0], u1 = vp[1], u2 = vp[2], u3 = vp[3];

    float p;
    p = __expf(st[0]); l += p; a0 = __builtin_fmaf(p, u0.x, a0); a1 = __builtin_fmaf(p, u0.y, a1);
    p = __expf(st[1]); l += p; a0 = __builtin_fmaf(p, u0.z, a0); a1 = __builtin_fmaf(p, u0.w, a1);
    p = __expf(st[2]); l += p; a0 = __builtin_fmaf(p, u1.x, a0); a1 = __builtin_fmaf(p, u1.y, a1);
    p = __expf(st[3]); l += p; a0 = __builtin_fmaf(p, u1.z, a0); a1 = __builtin_fmaf(p, u1.w, a1);
    p = __expf(st[4]); l += p; a0 = __builtin_fmaf(p, u2.x, a0); a1 = __builtin_fmaf(p, u2.y, a1);
    p = __expf(st[5]); l += p; a0 = __builtin_fmaf(p, u2.z, a0); a1 = __builtin_fmaf(p, u2.w, a1);
    p = __expf(st[6]); l += p; a0 = __builtin_fmaf(p, u3.x, a0); a1 = __builtin_fmaf(p, u3.y, a1);
    p = __expf(st[7]); l += p; a0 = __builtin_fmaf(p, u3.z, a0); a1 = __builtin_fmaf(p, u3.w, a1);
  }

  // Each query's key-space is split across lane pair (n, n+16): one combine.
  l  += __shfl_xor(l, 16);
  a0 += __shfl_xor(a0, 16);
  a1 += __shfl_xor(a1, 16);

  if (hi == 0) {
    const int t = q0 + l15;
    const int b = bh >> 5;
    const int h = bh & (kH - 1);
    const float inv = 1.0f / l;
    v2f o;
    o.x = a0 * inv;
    o.y = a1 * inv;
    *(v2f*)(Oo + (size_t)(b * kT + t) * kC + h * kD) = o;  // (B,T,C) layout
  }
}

// ---------------------------------------------------------------------------
// Kernel C: final projection y = O @ W_out^T, same WMMA tiling as kernel A.
// ---------------------------------------------------------------------------
__global__ __launch_bounds__(128) void out_proj_kernel(
    const float* __restrict__ Oi, const float* __restrict__ wout,
    float* __restrict__ y) {
  const int lane = threadIdx.x & 31;
  const int wv   = threadIdx.x >> 5;
  const int tile = blockIdx.x * 4 + wv;
  const int m0 = (tile >> 2) * 16;            // 4 column tiles
  const int n0 = (tile & 3) * 16;
  const int l15 = lane & 15;
  const int hi  = lane >> 4;

  const float* ap = Oi   + (m0 + l15) * kC + hi * 2;
  const float* bp = wout + (n0 + l15) * kC + hi * 2;

  v8f acc0 = {0.f, 0.f, 0.f, 0.f, 0.f, 0.f, 0.f, 0.f};
  v8f acc1 = {0.f, 0.f, 0.f, 0.f, 0.f, 0.f, 0.f, 0.f};
#pragma unroll
  for (int kk = 0; kk < kC; kk += 8) {
    v2f a0 = *(const v2f*)(ap + kk);
    v2f b0 = *(const v2f*)(bp + kk);
    v2f a1 = *(const v2f*)(ap + kk + 4);
    v2f b1 = *(const v2f*)(bp + kk + 4);
    acc0 = __builtin_amdgcn_wmma_f32_16x16x4_f32(false, a0, false, b0, (short)0,
                                                 acc0, false, false);
    acc1 = __builtin_amdgcn_wmma_f32_16x16x4_f32(false, a1, false, b1, (short)0,
                                                 acc1, false, false);
  }
#pragma unroll
  for (int r = 0; r < 8; ++r) {
    const int row = m0 + r + hi * 8;
    y[(size_t)row * kC + n0 + l15] = acc0[r] + acc1[r];
  }
}

// ---------------------------------------------------------------------------
extern "C" void kernel_launch(void* const* d_in, const int* in_sizes, int n_in,
                              void* d_out, int out_size, void* d_ws,
                              size_t ws_size, hipStream_t stream) {
  const float* x     = (const float*)d_in[0];
  const float* wqkv  = (const float*)d_in[1];
  const float* wout  = (const float*)d_in[2];
  const float* theta = (const float*)d_in[3];

  float* ws = (float*)d_ws;
  const size_t per = (size_t)kB * kH * kT * kD;  // 262144 floats = 1 MB
  float* Q  = ws;                 // [bh][t][d]
  float* KV = ws + per;           // [bh*2 + {K,V}][t][d]  (2 MB)
  float* O  = ws + 3 * per;       // (B,T,C)
  float* Y  = (float*)d_out;

  // A: 256 row-tiles x 12 col-tiles, 4 waves/block -> 768 blocks
  qkv_rope_kernel<<<(kTok / 16) * (kQKVC / 16) / 4, 128, 0, stream>>>(
      x, wqkv, theta, Q, KV);
  // B: 64 head-slices x 16 query-chunks, 8 waves/block -> 1024 blocks
  flash_attn_kernel<<<(kB * kH) * (kT / 16) / 8, 256, 0, stream>>>(Q, KV, O);
  // C: 256 row-tiles x 4 col-tiles, 4 waves/block -> 256 blocks
  out_proj_kernel<<<(kTok / 16) * (kC / 16) / 4, 128, 0, stream>>>(O, wout, Y);
}